// BPKDV4_24979529793866
// MI455X (gfx1250) — compile-verified
//
#include <hip/hip_runtime.h>
#include <math.h>

// Problem geometry (fixed by the reference): B=2, C=14, 96^3 volume.
#define NVOX   884736        // 96*96*96
#define PLANE  9216          // 96*96
#define ROWW   96
#define CH     2048          // voxels per block chunk
#define NBLK   (NVOX / CH)   // 432 chunks per batch
#define NCLS   14
#define NSLOT  112           // 2 types * 14 classes * 4 quantities

// Low 32 bits of a generic pointer to LDS == LDS byte offset (aperture layout:
// aperture base in the high 32 bits, LDS offset in the low 32 bits).
__device__ __forceinline__ unsigned int lds_lo32(const void* p) {
    return (unsigned int)(unsigned long long)p;
}

// Stage 8KB (2048 floats) global -> LDS with two async b128 loads per thread
// (per wave: 2 instructions, 512B each). ASYNCcnt += 2 per wave.
__device__ __forceinline__ void async_stage_8kb(const float* __restrict__ gsrc,
                                                float* lds, int t) {
    unsigned int l0 = lds_lo32(lds + t * 4);
    unsigned int l1 = l0 + 4096;
    int v0 = t * 16;
    int v1 = v0 + 4096;
    unsigned long long ga = (unsigned long long)gsrc;
    asm volatile("global_load_async_to_lds_b128 %0, %1, %2"
                 :: "v"(l0), "v"(v0), "s"(ga) : "memory");
    asm volatile("global_load_async_to_lds_b128 %0, %1, %2"
                 :: "v"(l1), "v"(v1), "s"(ga) : "memory");
}

__global__ __launch_bounds__(256) void bpkd_main(
        const float* __restrict__ S, const float* __restrict__ T,
        const int* __restrict__ gt, float* __restrict__ partials) {
    const int t = threadIdx.x;
    const int k = blockIdx.x;          // chunk within batch
    const int b = blockIdx.y;          // batch
    const int base = k * CH;           // first voxel of chunk

    __shared__ unsigned int maskw[CH];
    __shared__ float bufS[2][CH];
    __shared__ float bufT[2][CH];
    __shared__ float waccum[8][NSLOT]; // per-wave deterministic accumulators

    const float* Sb = S + (unsigned long long)(b * NCLS) * NVOX + base;
    const float* Tb = T + (unsigned long long)(b * NCLS) * NVOX + base;

    // Kick off async staging of class 1 into buffer 1 immediately so the DMA
    // overlaps the gt stencil below. (4 async ops outstanding per wave.)
    async_stage_8kb(Tb + 1 * NVOX, &bufT[1][0], t);
    async_stage_8kb(Sb + 1 * NVOX, &bufS[1][0], t);

    for (int i = t; i < 8 * NSLOT; i += 256) ((float*)waccum)[i] = 0.f;

    // Per-voxel mask words: bits[13:1] = edge_c, bit(16+g) = body (ero).
    const int* g0 = gt + b * NVOX;
    for (int i = 0; i < 8; ++i) {
        int v = base + i * 256 + t;
        int d = v / PLANE;
        int rem = v - d * PLANE;
        int h = rem / ROWW;
        int w = rem - h * ROWW;
        int g = g0[v];
        unsigned int dil = 1u << g;
        bool ero = (d > 0) & (d < 95) & (h > 0) & (h < 95) & (w > 0) & (w < 95);
        int n;
        if (w > 0)  { n = g0[v - 1];     dil |= 1u << n; ero &= (n == g); }
        if (w < 95) { n = g0[v + 1];     dil |= 1u << n; ero &= (n == g); }
        if (h > 0)  { n = g0[v - ROWW];  dil |= 1u << n; ero &= (n == g); }
        if (h < 95) { n = g0[v + ROWW];  dil |= 1u << n; ero &= (n == g); }
        if (d > 0)  { n = g0[v - PLANE]; dil |= 1u << n; ero &= (n == g); }
        if (d < 95) { n = g0[v + PLANE]; dil |= 1u << n; ero &= (n == g); }
        unsigned int edge = dil;
        if (ero) edge &= ~(1u << g);
        edge &= 0x3FFEu;                               // clear class 0
        unsigned int body = (ero && g != 0) ? (1u << (16 + g)) : 0u;
        maskw[i * 256 + t] = edge | body;
    }

    for (int c = 1; c <= 13; ++c) {
        const int cur = c & 1;
        if (c < 13) {
            // Prefetch next class into the other buffer, then wait until only
            // those 4 per-wave ops remain outstanding => current buffer done.
            async_stage_8kb(Tb + (c + 1) * NVOX, &bufT[cur ^ 1][0], t);
            async_stage_8kb(Sb + (c + 1) * NVOX, &bufS[cur ^ 1][0], t);
            asm volatile("s_wait_asynccnt 4" ::: "memory");
        } else {
            asm volatile("s_wait_asynccnt 0" ::: "memory");
        }
        __syncthreads();   // cross-wave visibility of staged data (+ masks on c==1)

        float et = 0.f, es = 0.f, ea = 0.f, ec = 0.f;   // edge sums
        float bt = 0.f, bs = 0.f, ba = 0.f, bc = 0.f;   // body sums
        const unsigned int ebit = 1u << c;
        const unsigned int bbit = 1u << (16 + c);
        #pragma unroll
        for (int i = 0; i < 8; ++i) {
            int v = i * 256 + t;                         // lane-contiguous: no bank conflicts
            unsigned int m = maskw[v];
            float Tv = bufT[cur][v];
            float Sv = bufS[cur][v];
            if (m & ebit) {
                float e1 = __expf(Tv), e2 = __expf(Sv);
                et += e1; es += e2; ea += e1 * (Tv - Sv); ec += 1.f;
            }
            if (m & bbit) {
                float e1 = __expf(Tv), e2 = __expf(Sv);
                bt += e1; bs += e2; ba += e1 * (Tv - Sv); bc += 1.f;
            }
        }
        // wave32 tree reduction
        #pragma unroll
        for (int o = 16; o > 0; o >>= 1) {
            et += __shfl_down(et, o); es += __shfl_down(es, o);
            ea += __shfl_down(ea, o); ec += __shfl_down(ec, o);
            bt += __shfl_down(bt, o); bs += __shfl_down(bs, o);
            ba += __shfl_down(ba, o); bc += __shfl_down(bc, o);
        }
        if ((t & 31) == 0) {                             // lane 0: private wave slot
            float* a = &waccum[t >> 5][0];
            int ce = c * 4, cb = 56 + c * 4;
            a[ce + 0] += et; a[ce + 1] += es; a[ce + 2] += ea; a[ce + 3] += ec;
            a[cb + 0] += bt; a[cb + 1] += bs; a[cb + 2] += ba; a[cb + 3] += bc;
        }
        __syncthreads();   // all waves done reading buf[cur] before it is refilled
    }

    if (t < NSLOT) {
        float s = 0.f;
        #pragma unroll
        for (int w = 0; w < 8; ++w) s += waccum[w][t];   // fixed order: deterministic
        partials[(unsigned long long)(b * NBLK + k) * NSLOT + t] = s;
    }
}

__global__ __launch_bounds__(256) void bpkd_final(
        const float* __restrict__ partials, float* __restrict__ out) {
    __shared__ double sh[2 * NSLOT];   // [b][slot]
    __shared__ double rl[56];
    const int t = threadIdx.x;
    if (t < 2 * NSLOT) {
        int b = t / NSLOT, i = t % NSLOT;
        double s = 0.0;
        const float* p = partials + (unsigned long long)b * NBLK * NSLOT + i;
        for (int k = 0; k < NBLK; ++k) s += (double)p[(unsigned long long)k * NSLOT];
        sh[t] = s;
    }
    __syncthreads();
    if (t < 56) {
        int type = t / 28, r = t % 28, b = r / 14, c = r % 14;
        int base = b * NSLOT + type * 56 + c * 4;
        double se = sh[base + 0], ss = sh[base + 1];
        double sa = sh[base + 2], cn = sh[base + 3];
        double rest = (double)NVOX - cn;                 // unmasked voxels: exp(0)=1 each
        double Zt = rest + se, Zs = rest + ss;
        rl[t] = sa / Zt + log(Zs) - log(Zt);
    }
    __syncthreads();
    if (t == 0) {
        double sE = 0.0, sB = 0.0;
        for (int i = 0; i < 28; ++i) sE += rl[i];
        for (int i = 28; i < 56; ++i) sB += rl[i];
        out[0] = (float)(sE * (500.0 / 56.0));   // 1/(B*C) * EDGE_W / B
        out[1] = (float)(sB * (200.0 / 56.0));   // 1/(B*C) * BODY_W / B
    }
}

extern "C" void kernel_launch(void* const* d_in, const int* in_sizes, int n_in,
                              void* d_out, int out_size, void* d_ws, size_t ws_size,
                              hipStream_t stream) {
    const float* S  = (const float*)d_in[0];   // preds_S [2,14,96,96,96] f32
    const float* T  = (const float*)d_in[1];   // preds_T [2,14,96,96,96] f32
    const int*   gt = (const int*)d_in[2];     // gt_labels [2,1,96,96,96] i32
    float* partials = (float*)d_ws;            // 2*432*112*4 = 387072 bytes
    (void)in_sizes; (void)n_in; (void)out_size; (void)ws_size;

    dim3 grid(NBLK, 2);
    bpkd_main<<<grid, 256, 0, stream>>>(S, T, gt, partials);
    bpkd_final<<<1, 256, 0, stream>>>(partials, (float*)d_out);
}